// GCN_69526930588080
// MI455X (gfx1250) — compile-verified
//
#include <hip/hip_runtime.h>

typedef float v2f __attribute__((ext_vector_type(2)));
typedef float v8f __attribute__((ext_vector_type(8)));

#define N_Sc 50000
#define N_Tc 50000
#define N_E  600000
#define D_INc 256
#define D_Hc  128
#define D_OUTc 64

// ---------------- degree / dinv ----------------
__global__ void k_init_deg(float* __restrict__ deg) {
    int i = blockIdx.x * blockDim.x + threadIdx.x;
    if (i < N_Tc) deg[i] = 1.0f;               // self loop
}

__global__ void k_deg_accum(const int* __restrict__ dst, float* __restrict__ deg) {
    int i = blockIdx.x * blockDim.x + threadIdx.x;
    if (i < N_E) atomicAdd(&deg[dst[i]], 1.0f);
}

__global__ void k_deg_to_dinv(float* __restrict__ deg) {
    int i = blockIdx.x * blockDim.x + threadIdx.x;
    if (i < N_Tc) deg[i] = rsqrtf(deg[i]);     // deg >= 1 always
}

// ---------------- fp32 WMMA GEMM: H = X @ W ----------------
// One wave computes a full 16-row x N strip of H, holding N/16 accumulator
// tiles so each A fragment is loaded once and reused across all column tiles.
// X rows [0,N_S) come from Xs, rows [N_S,N_S+N_T) from Xt; same split for
// Hs/Ht outputs. N_S is a multiple of 16, so a strip never straddles the
// boundary -> all selects are wave-uniform and EXEC stays all-ones.
template<int K, int N>
__global__ void k_gemm_wmma(const float* __restrict__ Xs, const float* __restrict__ Xt,
                            const float* __restrict__ W,
                            float* __restrict__ Hs, float* __restrict__ Ht,
                            int stripsTotal) {
    const int wave = threadIdx.x >> 5;
    const int lane = threadIdx.x & 31;
    const int strip = blockIdx.x * (blockDim.x >> 5) + wave;
    if (strip >= stripsTotal) return;          // wave-uniform exit

    constexpr int NT = N / 16;                 // column tiles in the strip
    const int tm = strip * 16;

    const int m15   = lane & 15;               // A row / B-D col within tile
    const int hi    = lane >> 4;               // 0: lanes 0-15, 1: lanes 16-31
    const int khalf = hi << 1;                 // K sub-offset 0 or 2

    // wave-uniform src/dst half selection
    const bool srcHalf = (tm < N_Sc);
    const float* xbase = srcHalf ? (Xs + (size_t)tm * K)
                                 : (Xt + (size_t)(tm - N_Sc) * K);
    const float* xrow  = xbase + (size_t)m15 * K;       // this lane's A row

    v8f acc[NT];
    #pragma unroll
    for (int j = 0; j < NT; ++j) acc[j] = (v8f){0.f,0.f,0.f,0.f,0.f,0.f,0.f,0.f};

    #pragma unroll 2
    for (int k = 0; k < K; k += 4) {
        const int k0 = k + khalf;
        const v2f a = *(const v2f*)(xrow + k0);         // one b64 load, reused NT times
        const float* w0 = W + (size_t)k0 * N + m15;
        const float* w1 = w0 + N;
        #pragma unroll
        for (int j = 0; j < NT; ++j) {
            v2f b;
            b.x = w0[j * 16];
            b.y = w1[j * 16];
            acc[j] = __builtin_amdgcn_wmma_f32_16x16x4_f32(
                         false, a, false, b, (short)0, acc[j], false, false);
        }
    }

    // store: VGPR i of tile j -> row tm+i+hi*8, col j*16+m15
    float* hbase = srcHalf ? (Hs + (size_t)tm * N)
                           : (Ht + (size_t)(tm - N_Sc) * N);
    const int roff = hi << 3;
    #pragma unroll
    for (int i = 0; i < 8; ++i) {
        float* hrow = hbase + (size_t)(i + roff) * N + m15;
        #pragma unroll
        for (int j = 0; j < NT; ++j) hrow[j * 16] = acc[j][i];
    }
}

// ---------------- aggregation helpers ----------------
// acc[t][f] = dinv[t] * Ht[t][f]
template<int D>
__global__ void k_init_acc(const float* __restrict__ Ht, const float* __restrict__ dinv,
                           float* __restrict__ acc) {
    int i = blockIdx.x * blockDim.x + threadIdx.x;
    if (i < N_Tc * D) acc[i] = dinv[i / D] * Ht[i];
}

// acc[dst[e]] += Hs[src[e]]  (float4 chunks, f32 global atomics)
template<int D>
__global__ void k_scatter(const int* __restrict__ src, const int* __restrict__ dst,
                          const float* __restrict__ Hs, float* __restrict__ acc) {
    constexpr int chunks = D / 4;
    int i = blockIdx.x * blockDim.x + threadIdx.x;
    if (i >= N_E * chunks) return;
    const int e = i / chunks;
    const int f = (i - e * chunks) * 4;
    const int s = src[e];
    const int t = dst[e];
    const float4 v = *(const float4*)(Hs + (size_t)s * D + f);
    float* p = acc + (size_t)t * D + f;
    atomicAdd(p + 0, v.x);
    atomicAdd(p + 1, v.y);
    atomicAdd(p + 2, v.z);
    atomicAdd(p + 3, v.w);
}

// out[t][f] = act(dinv[t] * acc[t][f])
template<int D, bool RELU>
__global__ void k_finalize_t(const float* __restrict__ acc, const float* __restrict__ dinv,
                             float* __restrict__ out) {
    int i = blockIdx.x * blockDim.x + threadIdx.x;
    if (i < N_Tc * D) {
        float v = dinv[i / D] * acc[i];
        if (RELU) v = fmaxf(v, 0.0f);
        out[i] = v;
    }
}

__global__ void k_relu_inplace(float* __restrict__ x, int n) {
    int i = blockIdx.x * blockDim.x + threadIdx.x;
    if (i < n) x[i] = fmaxf(x[i], 0.0f);
}

// ---------------- launcher ----------------
extern "C" void kernel_launch(void* const* d_in, const int* in_sizes, int n_in,
                              void* d_out, int out_size, void* d_ws, size_t ws_size,
                              hipStream_t stream) {
    const int*   edge = (const int*)  d_in[0];   // (2, E): [0..E)=src, [E..2E)=dst
    const float* x_s  = (const float*)d_in[1];   // (N_S, 256)
    const float* x_t  = (const float*)d_in[2];   // (N_T, 256)
    const float* W0   = (const float*)d_in[3];   // (256, 128)
    const float* W1   = (const float*)d_in[4];   // (128, 64)
    float* out = (float*)d_out;                  // [x_s (N_S,64) | x_t (N_T,64)]

    const int* src = edge;
    const int* dst = edge + N_E;

    // workspace layout (floats)
    float* ws   = (float*)d_ws;
    float* dinv = ws;                                        // N_T
    float* h1   = dinv + N_Tc;                               // (N_S+N_T) x 128
    float* acc1 = h1 + (size_t)(N_Sc + N_Tc) * D_Hc;         // N_T x 128
    float* h2t  = acc1 + (size_t)N_Tc * D_Hc;                // N_T x 64
    float* acc2 = h2t + (size_t)N_Tc * D_OUTc;               // N_T x 64

    const int B = 256;
    #define GRID(n) (((n) + B - 1) / B)

    // degrees -> dinv
    k_init_deg<<<GRID(N_Tc), B, 0, stream>>>(dinv);
    k_deg_accum<<<GRID(N_E), B, 0, stream>>>(dst, dinv);
    k_deg_to_dinv<<<GRID(N_Tc), B, 0, stream>>>(dinv);

    // ---- layer 1: h1 = [x_s;x_t] @ W0 ----
    {
        const int strips = (N_Sc + N_Tc) / 16;               // 6250
        const int blocks = (strips + 7) / 8;                 // 8 waves / block
        k_gemm_wmma<D_INc, D_Hc><<<blocks, B, 0, stream>>>(
            x_s, x_t, W0, h1, h1 + (size_t)N_Sc * D_Hc, strips);
    }
    // acc1 = dinv * h1_t ; acc1 += scatter(h1_s) ; x1_t = relu(dinv*acc1) in place
    k_init_acc<D_Hc><<<GRID(N_Tc * D_Hc), B, 0, stream>>>(
        h1 + (size_t)N_Sc * D_Hc, dinv, acc1);
    k_scatter<D_Hc><<<GRID(N_E * (D_Hc / 4)), B, 0, stream>>>(src, dst, h1, acc1);
    k_finalize_t<D_Hc, true><<<GRID(N_Tc * D_Hc), B, 0, stream>>>(acc1, dinv, acc1);
    // x1_s = relu(h1_s) in place
    k_relu_inplace<<<GRID(N_Sc * D_Hc), B, 0, stream>>>(h1, N_Sc * D_Hc);

    // ---- layer 2: h2 = [x1_s; x1_t] @ W1 ; src half straight into d_out ----
    {
        const int strips = (N_Sc + N_Tc) / 16;               // 6250
        const int blocks = (strips + 7) / 8;
        k_gemm_wmma<D_Hc, D_OUTc><<<blocks, B, 0, stream>>>(
            h1 /*x1_s*/, acc1 /*x1_t*/, W1, out /*h2_s -> final x_s*/, h2t, strips);
    }
    // acc2 = dinv * h2_t ; acc2 += scatter(h2_s = out[:N_S]) ; x_t = dinv*acc2 (no relu)
    k_init_acc<D_OUTc><<<GRID(N_Tc * D_OUTc), B, 0, stream>>>(h2t, dinv, acc2);
    k_scatter<D_OUTc><<<GRID(N_E * (D_OUTc / 4)), B, 0, stream>>>(src, dst, out, acc2);
    k_finalize_t<D_OUTc, false><<<GRID(N_Tc * D_OUTc), B, 0, stream>>>(
        acc2, dinv, out + (size_t)N_Sc * D_OUTc);

    #undef GRID
}